// AttnDecoderRNN_35150012350790
// MI455X (gfx1250) — compile-verified
//
#include <hip/hip_runtime.h>
#include <cmath>

// Problem sizes (fixed by the reference)
#define B_ 64
#define S_ 512
#define H_ 1024
#define E_ 1024
#define V_ 50257

typedef __attribute__((ext_vector_type(2))) float v2f;
typedef __attribute__((ext_vector_type(8))) float v8f;
typedef __attribute__((ext_vector_type(4))) int   v4i;

#ifndef __has_builtin
#define __has_builtin(x) 0
#endif

// D(16x16,f32) = A(16x4,f32) * B(4x16,f32) + C   — fp32 WMMA, wave32
__device__ __forceinline__ v8f wmma4(v2f a, v2f b, v8f c) {
    return __builtin_amdgcn_wmma_f32_16x16x4_f32(
        /*neg_a=*/false, a, /*neg_b=*/false, b,
        /*c_mod=*/(short)0, c, /*reuse_a=*/false, /*reuse_b=*/false);
}

__device__ __forceinline__ float sigmoidf_(float x) {
    return 1.0f / (1.0f + __expf(-x));
}

// --- CDNA5 async global->LDS copy (16B per lane, tracked by ASYNCcnt) ------
__device__ __forceinline__ void ldsasync_b128(const float* g, float* l) {
#if __has_builtin(__builtin_amdgcn_global_load_async_to_lds_b128)
    __builtin_amdgcn_global_load_async_to_lds_b128(
        (__attribute__((address_space(1))) v4i*)(void*)g,
        (__attribute__((address_space(3))) v4i*)l, 0, 0);
#else
    asm volatile("global_load_async_to_lds_b128 %0, %1, off"
                 :: "v"((unsigned)(unsigned long)
                        (__attribute__((address_space(3))) void*)l),
                    "v"(g)
                 : "memory");
#endif
}

__device__ __forceinline__ void wait_async0() {
#if __has_builtin(__builtin_amdgcn_s_wait_asynccnt)
    __builtin_amdgcn_s_wait_asynccnt(0);
#else
    asm volatile("s_wait_asynccnt 0x0" ::: "memory");
#endif
}

// ---------------------------------------------------------------------------
// 1) Embedding gather: x[b,:] = emb[idx[b],:]
// ---------------------------------------------------------------------------
__global__ __launch_bounds__(256) void k_embed(const int* __restrict__ idx,
                                               const float* __restrict__ emb,
                                               float* __restrict__ x) {
    const int b   = blockIdx.x;
    const long r  = (long)idx[b] * E_;
    for (int e = threadIdx.x; e < E_; e += 256)
        x[b * E_ + e] = emb[r + e];
}

// ---------------------------------------------------------------------------
// 2) Fused GRU cell via WMMA (6 gate GEMMs accumulated per wave, gate math
//    applied in-register on the WMMA accumulators).
// ---------------------------------------------------------------------------
__global__ __launch_bounds__(256) void k_gru(const float* __restrict__ X, int K,
                                             const float* __restrict__ Hprev,
                                             const float* __restrict__ Wih,
                                             const float* __restrict__ Whh,
                                             const float* __restrict__ bih,
                                             const float* __restrict__ bhh,
                                             float* __restrict__ Hnew) {
    const int lane = threadIdx.x & 31;
    const int w    = threadIdx.x >> 5;      // 0..7
    const int mt   = w & 3;                 // m-tile 0..3
    const int ns   = w >> 2;                // n-strip 0..1
    const int m0   = mt * 16;
    const int l16  = lane & 15;
    const int hi   = lane >> 4;
    const int j    = blockIdx.x * 32 + ns * 16 + l16;   // output column
    const int arow = m0 + l16;

    const float* WihR = Wih + (size_t)j            * K;
    const float* WihZ = Wih + (size_t)(H_ + j)     * K;
    const float* WihN = Wih + (size_t)(2 * H_ + j) * K;
    const float* WhhR = Whh + (size_t)j            * H_;
    const float* WhhZ = Whh + (size_t)(H_ + j)     * H_;
    const float* WhhN = Whh + (size_t)(2 * H_ + j) * H_;
    const float* Arow = X     + (size_t)arow * K;
    const float* Hrow = Hprev + (size_t)arow * H_;

    v8f air = {}, aiz = {}, ain = {}, ahr = {}, ahz = {}, ahn = {};

    for (int k0 = 0; k0 < K; k0 += 4) {
        const int ka = k0 + 2 * hi;          // A: contiguous pair per lane
        const int kb = k0 + hi;              // B: vgpr0 -> k / vgpr1 -> k+2
        v2f ax, ah, br, bz, bn, cr, cz, cn;
        ax.x = Arow[ka];     ax.y = Arow[ka + 1];
        ah.x = Hrow[ka];     ah.y = Hrow[ka + 1];
        br.x = WihR[kb];     br.y = WihR[kb + 2];
        bz.x = WihZ[kb];     bz.y = WihZ[kb + 2];
        bn.x = WihN[kb];     bn.y = WihN[kb + 2];
        cr.x = WhhR[kb];     cr.y = WhhR[kb + 2];
        cz.x = WhhZ[kb];     cz.y = WhhZ[kb + 2];
        cn.x = WhhN[kb];     cn.y = WhhN[kb + 2];
        air = wmma4(ax, br, air);
        aiz = wmma4(ax, bz, aiz);
        ain = wmma4(ax, bn, ain);
        ahr = wmma4(ah, cr, ahr);
        ahz = wmma4(ah, cz, ahz);
        ahn = wmma4(ah, cn, ahn);
    }

    const float bir = bih[j],          bhr = bhh[j];
    const float biz = bih[H_ + j],     bhz = bhh[H_ + j];
    const float bin = bih[2 * H_ + j], bhn = bhh[2 * H_ + j];

#pragma unroll
    for (int r = 0; r < 8; ++r) {
        const int row = m0 + r + 8 * hi;     // C/D layout: vgpr r, halves +8
        const float rg = sigmoidf_((air[r] + bir) + (ahr[r] + bhr));
        const float zg = sigmoidf_((aiz[r] + biz) + (ahz[r] + bhz));
        const float ng = tanhf((ain[r] + bin) + rg * (ahn[r] + bhn));
        const float hp = Hprev[(size_t)row * H_ + j];
        Hnew[(size_t)row * H_ + j] = (1.0f - zg) * ng + zg * hp;
    }
}

// ---------------------------------------------------------------------------
// 3) t = h1 @ Wa   (M=64, N=1024, K=1024)
// ---------------------------------------------------------------------------
__global__ __launch_bounds__(256) void k_tgemm(const float* __restrict__ h1,
                                               const float* __restrict__ Wa,
                                               float* __restrict__ t) {
    const int lane = threadIdx.x & 31, w = threadIdx.x >> 5;
    const int l16 = lane & 15, hi = lane >> 4;
    for (int job = w; job < 16; job += 8) {
        const int m0 = (job & 3) * 16;
        const int n  = blockIdx.x * 64 + (job >> 2) * 16 + l16;
        const float* Arow = h1 + (size_t)(m0 + l16) * H_;
        v8f acc = {};
        for (int k0 = 0; k0 < H_; k0 += 4) {
            const int ka = k0 + 2 * hi;
            v2f a, b;
            a.x = Arow[ka];                       a.y = Arow[ka + 1];
            b.x = Wa[(size_t)(k0 + hi) * H_ + n]; b.y = Wa[(size_t)(k0 + 2 + hi) * H_ + n];
            acc = wmma4(a, b, acc);
        }
#pragma unroll
        for (int r = 0; r < 8; ++r)
            t[(size_t)(m0 + r + 8 * hi) * H_ + n] = acc[r];
    }
}

// ---------------------------------------------------------------------------
// 4) Fused attention per batch row b (score = (h1@Wa).enc + h1.ba, softmax,
//    context) — LDS + wave shuffles.
// ---------------------------------------------------------------------------
__global__ __launch_bounds__(512) void k_attn(const float* __restrict__ t,
                                              const float* __restrict__ h1,
                                              const float* __restrict__ ba,
                                              const float* __restrict__ enc,
                                              float* __restrict__ attnw,
                                              float* __restrict__ context) {
    __shared__ float sT[H_];
    __shared__ float sScore[S_];
    __shared__ float sRed[16];
    __shared__ float sBC;

    const int b = blockIdx.x;
    const int tid = threadIdx.x, lane = tid & 31, w = tid >> 5;   // 16 waves

    float cpart = 0.0f;
    for (int h = tid; h < H_; h += 512) {
        const float tv = t[(size_t)b * H_ + h];
        sT[h] = tv;
        cpart += h1[(size_t)b * H_ + h] * ba[h];
    }
    for (int off = 16; off; off >>= 1) cpart += __shfl_down(cpart, off, 32);
    if (lane == 0) sRed[w] = cpart;
    __syncthreads();
    if (tid == 0) { float s = 0.f; for (int i = 0; i < 16; ++i) s += sRed[i]; sBC = s; }
    __syncthreads();
    const float cb = sBC;

    for (int s = w; s < S_; s += 16) {
        const float* ev = enc + ((size_t)s * B_ + b) * H_;
        float d = 0.0f;
        for (int h = lane; h < H_; h += 32) d += sT[h] * ev[h];
        for (int off = 16; off; off >>= 1) d += __shfl_down(d, off, 32);
        if (lane == 0) sScore[s] = d + cb;
    }
    __syncthreads();

    const float val = sScore[tid];
    float mx = val;
    for (int off = 16; off; off >>= 1) mx = fmaxf(mx, __shfl_xor(mx, off, 32));
    if (lane == 0) sRed[w] = mx;
    __syncthreads();
    if (tid == 0) { float m = sRed[0]; for (int i = 1; i < 16; ++i) m = fmaxf(m, sRed[i]); sBC = m; }
    __syncthreads();
    const float ex = __expf(val - sBC);
    float sm = ex;
    for (int off = 16; off; off >>= 1) sm += __shfl_xor(sm, off, 32);
    if (lane == 0) sRed[w] = sm;
    __syncthreads();
    if (tid == 0) { float s = 0.f; for (int i = 0; i < 16; ++i) s += sRed[i]; sBC = 1.0f / s; }
    __syncthreads();
    const float wgt = ex * sBC;
    sScore[tid] = wgt;
    attnw[(size_t)b * S_ + tid] = wgt;
    __syncthreads();

    for (int h = tid; h < H_; h += 512) {
        float acc = 0.0f;
        for (int s = 0; s < S_; ++s)
            acc += sScore[s] * enc[((size_t)s * B_ + b) * H_ + h];
        context[(size_t)b * H_ + h] = acc;
    }
}

// ---------------------------------------------------------------------------
// 5) concat_output = tanh([h1|ctx] @ Wc^T + bc)   (M=64, N=1024, K=2048)
// ---------------------------------------------------------------------------
__global__ __launch_bounds__(256) void k_concat(const float* __restrict__ h1,
                                                const float* __restrict__ ctx,
                                                const float* __restrict__ Wc,
                                                const float* __restrict__ bc,
                                                float* __restrict__ out) {
    const int lane = threadIdx.x & 31, w = threadIdx.x >> 5;
    const int l16 = lane & 15, hi = lane >> 4;
    for (int job = w; job < 16; job += 8) {
        const int m0 = (job & 3) * 16;
        const int n  = blockIdx.x * 64 + (job >> 2) * 16 + l16;
        const int ar = m0 + l16;
        const float* Wrow = Wc + (size_t)n * (2 * H_);
        v8f acc = {};
        for (int k0 = 0; k0 < 2 * H_; k0 += 4) {
            const float* A = (k0 < H_) ? h1 : ctx;
            const int kk = (k0 < H_) ? k0 : k0 - H_;
            const int ka = kk + 2 * hi;
            v2f a, b;
            a.x = A[(size_t)ar * H_ + ka]; a.y = A[(size_t)ar * H_ + ka + 1];
            b.x = Wrow[k0 + hi];           b.y = Wrow[k0 + 2 + hi];
            acc = wmma4(a, b, acc);
        }
        const float bias = bc[n];
#pragma unroll
        for (int r = 0; r < 8; ++r)
            out[(size_t)(m0 + r + 8 * hi) * H_ + n] = tanhf(acc[r] + bias);
    }
}

// ---------------------------------------------------------------------------
// 6) output = concat @ Wo^T + bo   (M=64, N=50257 ragged, K=1024)
//    Double-buffered ASYNC global->LDS staging of Wo and A tiles; WMMA
//    operands come from conflict-free LDS (padded stride), so the 206 MB Wo
//    HBM stream runs ahead of compute on the async engine.
// ---------------------------------------------------------------------------
#define KC_   32                 // K-chunk (floats)
#define WPAD_ 36                 // padded LDS row stride (floats)

__global__ __launch_bounds__(256) void k_out(const float* __restrict__ A,
                                             const float* __restrict__ Wo,
                                             const float* __restrict__ bo,
                                             float* __restrict__ out) {
    __shared__ float sW[2][64 * WPAD_];   // Wo tile: 64 vocab rows x KC_
    __shared__ float sA[2][64 * WPAD_];   // A  tile: 64 batch rows x KC_

    const int tid  = threadIdx.x;
    const int lane = tid & 31, w = tid >> 5;
    const int l16  = lane & 15, hi = lane >> 4;
    const int n0    = blockIdx.x * 64;
    const int strip = (w & 3) * 16;            // 16 vocab rows per n-strip
    const int mbase = (w >> 2) * 32;           // m-tiles {0,1} or {2,3}
    const int n     = n0 + strip + l16;
    const bool valid = (n < V_);

    // stage one K-chunk of Wo and A into LDS buffer `buf` (async, 16B/lane)
    auto stage = [&](int kc, int buf) {
#pragma unroll
        for (int i = 0; i < 2; ++i) {
            const int f = tid + 256 * i;       // float4 index 0..511
            const int r = f >> 3;              // row 0..63
            const int c = (f & 7) << 2;        // col 0..28, step 4
            int gr = n0 + r; if (gr >= V_) gr = 0;   // clamp ragged tail
            ldsasync_b128(Wo + (size_t)gr * H_ + kc + c,
                          &sW[buf][r * WPAD_ + c]);
            ldsasync_b128(A + (size_t)r * H_ + kc + c,
                          &sA[buf][r * WPAD_ + c]);
        }
    };

    stage(0, 0);
    wait_async0();
    __syncthreads();

    v8f acc0 = {}, acc1 = {};
    for (int kc = 0; kc < H_; kc += KC_) {
        const int buf = (kc / KC_) & 1;
        if (kc + KC_ < H_) stage(kc + KC_, buf ^ 1);   // prefetch next chunk

        const float* Wl  = &sW[buf][(strip + l16) * WPAD_];
        const float* Al0 = &sA[buf][(mbase + l16) * WPAD_];
        const float* Al1 = Al0 + 16 * WPAD_;
#pragma unroll
        for (int kk = 0; kk < KC_; kk += 4) {
            const int ka = kk + 2 * hi;
            v2f b, a0, a1;
            b.x  = Wl[kk + hi];  b.y  = Wl[kk + 2 + hi];
            a0.x = Al0[ka];      a0.y = Al0[ka + 1];
            a1.x = Al1[ka];      a1.y = Al1[ka + 1];
            acc0 = wmma4(a0, b, acc0);
            acc1 = wmma4(a1, b, acc1);
        }
        wait_async0();       // prefetch landed
        __syncthreads();     // everyone done with buf before it is re-staged
    }

    if (valid) {
        const float bias = bo[n];
#pragma unroll
        for (int r = 0; r < 8; ++r) {
            const int row = mbase + r + 8 * hi;
            out[(size_t)row * V_ + n]        = acc0[r] + bias;
            out[(size_t)(row + 16) * V_ + n] = acc1[r] + bias;
        }
    }
}

// ---------------------------------------------------------------------------
extern "C" void kernel_launch(void* const* d_in, const int* in_sizes, int n_in,
                              void* d_out, int out_size, void* d_ws, size_t ws_size,
                              hipStream_t stream) {
    const int*   input_seq   = (const int*)  d_in[0];
    const float* last_hidden = (const float*)d_in[1];
    const float* enc         = (const float*)d_in[2];
    const float* emb         = (const float*)d_in[3];
    const float* Wih0        = (const float*)d_in[4];
    const float* Whh0        = (const float*)d_in[5];
    const float* bih0        = (const float*)d_in[6];
    const float* bhh0        = (const float*)d_in[7];
    const float* Wih1        = (const float*)d_in[8];
    const float* Whh1        = (const float*)d_in[9];
    const float* bih1        = (const float*)d_in[10];
    const float* bhh1        = (const float*)d_in[11];
    const float* Wa          = (const float*)d_in[12];
    const float* ba          = (const float*)d_in[13];
    const float* Wc          = (const float*)d_in[14];
    const float* bc          = (const float*)d_in[15];
    const float* Wo          = (const float*)d_in[16];
    const float* bo          = (const float*)d_in[17];

    // d_out = [output BxV | hidden LxBxH | attn Bx1xS]
    float* out   = (float*)d_out;
    float* h0    = out + (size_t)B_ * V_;
    float* h1    = h0 + (size_t)B_ * H_;
    float* attnw = h1 + (size_t)B_ * H_;

    // workspace: x_emb, t, context, concat (64x1024 each)
    float* ws    = (float*)d_ws;
    float* x_emb = ws;
    float* t     = ws + 1 * B_ * H_;
    float* ctx   = ws + 2 * B_ * H_;
    float* cat   = ws + 3 * B_ * H_;

    k_embed <<<B_, 256, 0, stream>>>(input_seq, emb, x_emb);
    k_gru   <<<H_ / 32, 256, 0, stream>>>(x_emb, E_, last_hidden,
                                          Wih0, Whh0, bih0, bhh0, h0);
    k_gru   <<<H_ / 32, 256, 0, stream>>>(h0, H_, last_hidden + B_ * H_,
                                          Wih1, Whh1, bih1, bhh1, h1);
    k_tgemm <<<H_ / 64, 256, 0, stream>>>(h1, Wa, t);
    k_attn  <<<B_, 512, 0, stream>>>(t, h1, ba, enc, attnw, ctx);
    k_concat<<<H_ / 64, 256, 0, stream>>>(h1, ctx, Wc, bc, cat);
    k_out   <<<(V_ + 63) / 64, 256, 0, stream>>>(cat, Wo, bo, out);
}